// MedianConvolution_27496380629011
// MI455X (gfx1250) — compile-verified
//
#include <hip/hip_runtime.h>
#include <hip/hip_bf16.h>

typedef __attribute__((ext_vector_type(2))) float v2f;
typedef __attribute__((ext_vector_type(8))) float v8f;

#define NNODES 50000
#define FDIM   256
#define UNITS  64
#define KNBR   32

// ---------------------------------------------------------------------------
// Kernel 1: h = x @ kernel  using V_WMMA_F32_16X16X4_F32 (exact fp32)
// Block: 256 threads (8 waves). Each wave owns one 16-row M-tile and computes
// the full 16x64 output slab (4 N-tiles). B (256x64 = 64KB) staged in LDS.
// ---------------------------------------------------------------------------
__global__ void __launch_bounds__(256)
mc_gemm_wmma(const float* __restrict__ x,
             const float* __restrict__ kern,
             float* __restrict__ h)
{
    __shared__ float ldsK[FDIM * UNITS];   // 64 KB (WGP has 320 KB)

    // cooperative coalesced load of kernel matrix into LDS
    {
        const float4* src = reinterpret_cast<const float4*>(kern);
        float4*       dst = reinterpret_cast<float4*>(ldsK);
        #pragma unroll
        for (int i = 0; i < (FDIM * UNITS / 4) / 256; ++i)
            dst[threadIdx.x + i * 256] = src[threadIdx.x + i * 256];
    }
    __syncthreads();

    const int wave = threadIdx.x >> 5;
    const int lane = threadIdx.x & 31;
    const int half = lane >> 4;      // lane group 0 or 1
    const int lrow = lane & 15;

    const int mtile = blockIdx.x * 8 + wave;          // 3125 M-tiles total
    if (mtile >= NNODES / 16) return;                 // wave-uniform exit
    const int m0 = mtile * 16;

    v8f acc[4] = {};  // 4 N-tiles of 16x16 f32 accumulators

    const float* arow = x + (size_t)(m0 + lrow) * FDIM;

    #pragma unroll 4
    for (int k0 = 0; k0 < FDIM; k0 += 4) {
        // A fragment (16x4 f32): lane group half holds K = 2*half + {0,1}
        const int ka = k0 + 2 * half;
        v2f a = *reinterpret_cast<const v2f*>(arow + ka);  // 8B aligned

        #pragma unroll
        for (int nt = 0; nt < 4; ++nt) {
            // B fragment (4x16 f32): same K striping, N = lrow
            v2f b;
            b.x = ldsK[(ka + 0) * UNITS + nt * 16 + lrow];
            b.y = ldsK[(ka + 1) * UNITS + nt * 16 + lrow];
            acc[nt] = __builtin_amdgcn_wmma_f32_16x16x4_f32(
                false, a, false, b, (short)0, acc[nt], false, false);
        }
    }

    // C/D layout: VGPR g -> row m0 + g + 8*half, col nt*16 + lrow
    #pragma unroll
    for (int nt = 0; nt < 4; ++nt) {
        #pragma unroll
        for (int g = 0; g < 8; ++g) {
            h[(size_t)(m0 + g + 8 * half) * UNITS + nt * 16 + lrow] = acc[nt][g];
        }
    }
}

// ---------------------------------------------------------------------------
// Kernel 2: gather K=32 neighbor messages per (node, unit), exact median via
// fully unrolled 32-input bitonic network held in VGPRs.
// Block: 256 threads = 4 nodes x 64 units. Gather of h[nbr][0..63] is a
// coalesced 256B line per (node,k) -> L2-resident (h is 12.8MB << 192MB L2).
// ---------------------------------------------------------------------------
__global__ void __launch_bounds__(256)
mc_median(const int* __restrict__ neighbors,   // [NNODES, 32] int32
          const float* __restrict__ h,         // [NNODES, 64]
          float* __restrict__ out)             // [NNODES, 64]
{
    __shared__ int nbr[4 * KNBR];

    // stage the 4 nodes' neighbor lists once (each is read by 64 threads)
    if (threadIdx.x < 4 * KNBR)
        nbr[threadIdx.x] = neighbors[(size_t)blockIdx.x * (4 * KNBR) + threadIdx.x];
    __syncthreads();

    const int nloc = threadIdx.x >> 6;          // node within block (0..3)
    const int u    = threadIdx.x & 63;          // unit
    const int node = blockIdx.x * 4 + nloc;     // 50000 = 4 * 12500 exactly

    float v[KNBR];
    #pragma unroll
    for (int k = 0; k < KNBR; ++k) {
        const int idx = nbr[nloc * KNBR + k];
        v[k] = h[(size_t)idx * UNITS + u];
    }

    // bitonic sort, ascending; constant indices -> stays in registers
    #pragma unroll
    for (int size = 2; size <= KNBR; size <<= 1) {
        #pragma unroll
        for (int stride = size >> 1; stride > 0; stride >>= 1) {
            #pragma unroll
            for (int i = 0; i < KNBR; ++i) {
                const int j = i ^ stride;
                if (j > i) {
                    const bool up = ((i & size) == 0);
                    const float a = v[i], b = v[j];
                    const float lo = fminf(a, b);
                    const float hi = fmaxf(a, b);
                    v[i] = up ? lo : hi;
                    v[j] = up ? hi : lo;
                }
            }
        }
    }

    out[(size_t)node * UNITS + u] = (v[KNBR / 2 - 1] + v[KNBR / 2]) * 0.5f;
}

extern "C" void kernel_launch(void* const* d_in, const int* in_sizes, int n_in,
                              void* d_out, int out_size, void* d_ws, size_t ws_size,
                              hipStream_t stream)
{
    (void)in_sizes; (void)n_in; (void)out_size; (void)ws_size;

    const float* x    = (const float*)d_in[0];
    const int*   nbrs = (const int*)d_in[1];     // jax x64-off: int32
    const float* kern = (const float*)d_in[2];
    float*       out  = (float*)d_out;
    float*       h    = (float*)d_ws;            // [50000, 64] fp32 = 12.8 MB

    // GEMM: 3125 M-tiles, 8 waves (tiles) per 256-thread block
    const int gemm_blocks = (NNODES / 16 + 7) / 8;   // 391
    mc_gemm_wmma<<<gemm_blocks, 256, 0, stream>>>(x, kern, h);

    // Median: 4 nodes per block
    mc_median<<<NNODES / 4, 256, 0, stream>>>(nbrs, h, out);
}